// E3DiffusionProcess_12962211299503
// MI455X (gfx1250) — compile-verified
//
#include <hip/hip_runtime.h>
#include <math.h>

#define NUM_GRAPHS 8192

typedef unsigned int v4u __attribute__((ext_vector_type(4)));
typedef int          v8i __attribute__((ext_vector_type(8)));
typedef int          v4i __attribute__((ext_vector_type(4)));

// ---------------------------------------------------------------------------
// ws layout (floats): [0]=alpha_t, [1]=sigma_t, [8 .. 8+3*NUM_GRAPHS) = mean
// ---------------------------------------------------------------------------

// Kernel A: polynomial_alpha_schedule()[t] -> alpha_t, sigma_t (f32-faithful).
__global__ void sched_kernel(const int* __restrict__ t_ptr,
                             float* __restrict__ ws) {
    if (blockIdx.x != 0 || threadIdx.x != 0) return;
    const int t = *t_ptr;            // 0..1000
    float prev = 1.0f;               // padded[0] = 1
    float cum  = 1.0f;
    for (int i = 0; i <= t; ++i) {
        float r  = (float)i / 1000.0f;         // x / T_STEPS
        float a2 = 1.0f - r * r * r;           // 1 - (x/T)^3
        a2 = a2 * a2;                          // (...)^2
        float step = a2 / prev;
        step = fminf(fmaxf(step, 0.001f), 1.0f);
        cum *= step;
        prev = a2;
    }
    float alpha = (1.0f - 2.0f * 1e-4f) * cum + 1e-4f;
    ws[0] = alpha;
    ws[1] = sqrtf(fmaxf(1.0f - alpha * alpha, 0.0f));
}

// Kernel B: per-graph mean of noise. batch_index is sorted, so segment g is
// the contiguous range [lower_bound(g), lower_bound(g+1)) -- no atomics.
// One block (8 wave32s) per graph. The segment is a contiguous 1-D tile, so
// wave 0 DMAs it into LDS with the Tensor Data Mover (tensor_load_to_lds,
// TENSORcnt) and the block reduces out of LDS. Oversized segments (or a
// toolchain without the builtin) fall back to direct global loads.
#define SEG_LDS_ROWS 2048   // 2048 rows * 12B = 24 KB LDS tile

__global__ void __launch_bounds__(256)
seg_mean_kernel(const float* __restrict__ noise,
                const int* __restrict__ idx,
                float* __restrict__ mean,
                int n) {
    __shared__ int sbounds[2];
    __shared__ float red[3][8];
#if __has_builtin(__builtin_amdgcn_tensor_load_to_lds)
    __shared__ float buf[3 * SEG_LDS_ROWS];
#endif

    const int g = blockIdx.x;
    if (threadIdx.x < 2) {
        const int target = g + (int)threadIdx.x;
        int lo = 0, hi = n;
        while (lo < hi) {
            int mid = (lo + hi) >> 1;
            if (idx[mid] < target) lo = mid + 1; else hi = mid;
        }
        sbounds[threadIdx.x] = lo;
    }
    __syncthreads();
    const int s = sbounds[0];
    const int e = sbounds[1];
    const int rows = e - s;

    float ax = 0.0f, ay = 0.0f, az = 0.0f;

#if __has_builtin(__builtin_amdgcn_tensor_load_to_lds)
    if (rows > 0 && rows <= SEG_LDS_ROWS) {
        // --- TDM path: DMA the whole segment (3*rows f32) into LDS. ---
        if (threadIdx.x < 32) {   // one wave issues the descriptor
            const unsigned long long gaddr =
                (unsigned long long)(const void*)(noise + 3 * (size_t)s);
            // Flat-LDS aperture: low 32 bits of a generic LDS pointer are the
            // LDS byte offset -> usable directly as D#.lds_addr.
            const unsigned lds_addr = (unsigned)(unsigned long long)(const void*)buf;
            const unsigned numFlat  = 3u * (unsigned)rows;   // elements (4B each)

            // D# group 0: count=1 | lds_addr | global_addr[56:0] | type=2
            v4u g0;
            g0.x = 1u;                                   // count=1, no gather
            g0.y = lds_addr;                             // bits 63:32
            g0.z = (unsigned)gaddr;                      // bits 95:64
            g0.w = (unsigned)((gaddr >> 32) & 0x1FFFFFFu) | (2u << 30); // type=2

            // D# group 1: data_size=4B; 1-D tile: tensor_dim0=tile_dim0=numFlat
            v8i g1;
            g1.s0 = 0x20000;                                  // data_size=2 (4B)
            g1.s1 = (int)((numFlat & 0xFFFFu) << 16);         // tensor_dim0[15:0]
            g1.s2 = (int)(((numFlat >> 16) & 0xFFFFu)         // tensor_dim0[31:16]
                          | (1u << 16));                      // tensor_dim1 = 1
            g1.s3 = (int)((numFlat & 0xFFFFu) << 16);         // tile_dim0 = numFlat
            g1.s4 = 0;                                        // tile_dim1/2 unused
            g1.s5 = (int)numFlat;                             // tensor_dim0_stride
            g1.s6 = 0;
            g1.s7 = 0;

            v4i g2 = {0, 0, 0, 0};
            v4i g3 = {0, 0, 0, 0};
#if defined(__clang_major__) && (__clang_major__ >= 23)
            v8i g4 = {0, 0, 0, 0, 0, 0, 0, 0};
            __builtin_amdgcn_tensor_load_to_lds(g0, g1, g2, g3, g4, 0);
#else
            __builtin_amdgcn_tensor_load_to_lds(g0, g1, g2, g3, 0);
#endif
#if __has_builtin(__builtin_amdgcn_s_wait_tensorcnt)
            __builtin_amdgcn_s_wait_tensorcnt(0);
#else
            asm volatile("s_wait_tensorcnt 0x0" ::: "memory");
#endif
        }
        __syncthreads();   // LDS tile now valid for all 8 waves

        for (int r = (int)threadIdx.x; r < rows; r += (int)blockDim.x) {
            ax += buf[3 * r + 0];
            ay += buf[3 * r + 1];
            az += buf[3 * r + 2];
        }
    } else if (rows > 0)
#endif
    {
        // --- Fallback: direct global streaming with gfx1250 prefetch. ---
        for (int i = s + (int)threadIdx.x; i < e; i += (int)blockDim.x) {
            __builtin_prefetch(&noise[3 * (i + 2 * (int)blockDim.x)], 0, 0);
            ax += noise[3 * i + 0];
            ay += noise[3 * i + 1];
            az += noise[3 * i + 2];
        }
    }

    // wave32 reduction, then cross-wave via LDS
    for (int off = 16; off > 0; off >>= 1) {
        ax += __shfl_down(ax, off, 32);
        ay += __shfl_down(ay, off, 32);
        az += __shfl_down(az, off, 32);
    }
    const int wave = threadIdx.x >> 5;
    const int lane = threadIdx.x & 31;
    if (lane == 0) { red[0][wave] = ax; red[1][wave] = ay; red[2][wave] = az; }
    __syncthreads();

    if (threadIdx.x == 0) {
        float sx = 0.0f, sy = 0.0f, sz = 0.0f;
        for (int w = 0; w < 8; ++w) { sx += red[0][w]; sy += red[1][w]; sz += red[2][w]; }
        const float d = fmaxf((float)rows, 1.0f);   // matches max(cnts, 1.0)
        mean[3 * g + 0] = sx / d;
        mean[3 * g + 1] = sy / d;
        mean[3 * g + 2] = sz / d;
    }
}

// Kernel C: noise_c = noise - mean[idx];  z_after = alpha*z + sigma*noise_c.
// z streamed once -> NT loads; outputs never re-read -> NT stores; noise
// stays RT so it hits the 192MB L2 warmed by kernel B.
__global__ void __launch_bounds__(256)
diffuse_kernel(const float* __restrict__ z,
               const float* __restrict__ noise,
               const int* __restrict__ idx,
               const float* __restrict__ ws,
               const float* __restrict__ mean,
               float* __restrict__ out,
               int n) {
    const int i = blockIdx.x * blockDim.x + threadIdx.x;
    if (i >= n) return;

    const float alpha = ws[0];
    const float sigma = ws[1];

    const int g = idx[i];
    const float mx = mean[3 * g + 0];
    const float my = mean[3 * g + 1];
    const float mz = mean[3 * g + 2];

    const float ncx = noise[3 * i + 0] - mx;
    const float ncy = noise[3 * i + 1] - my;
    const float ncz = noise[3 * i + 2] - mz;

    const float zx = __builtin_nontemporal_load(&z[3 * i + 0]);
    const float zy = __builtin_nontemporal_load(&z[3 * i + 1]);
    const float zz = __builtin_nontemporal_load(&z[3 * i + 2]);

    float* za_out = out;                       // z_after: first N*3 floats
    float* nc_out = out + (size_t)3 * n;       // noise_c: next N*3 floats

    __builtin_nontemporal_store(fmaf(alpha, zx, sigma * ncx), &za_out[3 * i + 0]);
    __builtin_nontemporal_store(fmaf(alpha, zy, sigma * ncy), &za_out[3 * i + 1]);
    __builtin_nontemporal_store(fmaf(alpha, zz, sigma * ncz), &za_out[3 * i + 2]);
    __builtin_nontemporal_store(ncx, &nc_out[3 * i + 0]);
    __builtin_nontemporal_store(ncy, &nc_out[3 * i + 1]);
    __builtin_nontemporal_store(ncz, &nc_out[3 * i + 2]);
}

extern "C" void kernel_launch(void* const* d_in, const int* in_sizes, int n_in,
                              void* d_out, int out_size, void* d_ws, size_t ws_size,
                              hipStream_t stream) {
    const float* z     = (const float*)d_in[0];   // (N,3) f32
    const float* noise = (const float*)d_in[1];   // (N,3) f32
    const int*   idx   = (const int*)d_in[2];     // (N,)  i32, sorted
    const int*   t     = (const int*)d_in[3];     // scalar i32

    const int n = in_sizes[2];                    // N nodes

    float* ws   = (float*)d_ws;
    float* mean = ws + 8;                         // 32B-aligned scratch region
    float* out  = (float*)d_out;

    sched_kernel<<<1, 32, 0, stream>>>(t, ws);
    seg_mean_kernel<<<NUM_GRAPHS, 256, 0, stream>>>(noise, idx, mean, n);
    diffuse_kernel<<<(n + 255) / 256, 256, 0, stream>>>(z, noise, idx, ws, mean, out, n);
}